// Focal_Reg_Loss_68341519614014
// MI455X (gfx1250) — compile-verified
//
#include <hip/hip_runtime.h>
#include <hip/hip_bf16.h>

typedef __attribute__((ext_vector_type(2))) float v2f;
typedef __attribute__((ext_vector_type(8))) float v8f;

// ---------- math helpers ----------

__device__ __forceinline__ float softplus_f(float z) {
  // log(1 + e^z), stable for large z
  return (z > 20.0f) ? z : log1pf(expf(z));
}

// focal(t, x) with ALPHA=0.25, GAMMA=2:
//  t==1: 0.25*(1-c)^2 * (-log c)      = 0.25*(1-c)^2 * softplus(-x)
//  t==0: 0.75*c^2     * (-log(1-c))   = 0.75*c^2     * softplus(x)
__device__ __forceinline__ float focal_elem(bool tpos, float x) {
  float c = 1.0f / (1.0f + expf(-x));
  float u = 1.0f - c;
  float fp = 0.25f * u * u * softplus_f(-x);
  float fn = 0.75f * c * c * softplus_f(x);
  return tpos ? fp : fn;
}

__device__ __forceinline__ float smooth_l1(float d) {
  return (d <= (1.0f / 9.0f)) ? 4.5f * d * d : d - (1.0f / 18.0f);
}

// ---------- WMMA-based wave32 reduction ----------
// A = all-ones 16x4 => D[m][n] = sum_k B[k][n]. Every row of D is the
// column-sum vector; one D VGPR holds two full rows across 32 lanes, so
// summing c[0] over all 32 lanes = 2 * (sum of all B elements) = 2*sum(p).
// Correct regardless of the exact (k,n)<->lane mapping of B.
// EXEC must be all 1s at the call site (no divergent early exits).
__device__ __forceinline__ float wave_reduce_wmma(float p) {
  v2f a; a[0] = 1.0f; a[1] = 1.0f;   // all 64 A elements = 1.0
  v2f b; b[0] = p;    b[1] = 0.0f;   // 32 partials + 32 zeros
  v8f c = {};
  c = __builtin_amdgcn_wmma_f32_16x16x4_f32(
      /*neg_a=*/false, a, /*neg_b=*/false, b,
      /*c_mod=*/(short)0, c, /*reuse_a=*/false, /*reuse_b=*/false);
  float v = c[0];
  v += __shfl_xor(v, 16, 32);
  v += __shfl_xor(v, 8, 32);
  v += __shfl_xor(v, 4, 32);
  v += __shfl_xor(v, 2, 32);
  v += __shfl_xor(v, 1, 32);
  return 0.5f * v;
}

// ---------- kernels ----------

__global__ void zero_ws_kernel(float* ws, int n) {
  int i = blockIdx.x * blockDim.x + threadIdx.x;
  if (i < n) ws[i] = 0.0f;
}

// Streaming pass: per-anchor IoU -> pos -> focal(pos, box_dist); accumulates
// focal sum (ws[b]) and pos count (ws[B+b]) per image.
__global__ void bbox_focal_kernel(const float* __restrict__ anchors,
                                  const float* __restrict__ annotations,
                                  const float* __restrict__ box_dist,
                                  float* __restrict__ ws, int A, int B) {
  const int b = blockIdx.y;
  const float* ann = annotations + (size_t)b * 7;
  const float bx1 = ann[0], by1 = ann[1], bx2 = ann[2], by2 = ann[3];
  const float area_b = (bx2 - bx1) * (by2 - by1);
  const float* bd = box_dist + (size_t)b * A;

  float fsum = 0.0f, psum = 0.0f;
  const int stride = gridDim.x * blockDim.x;
  for (int i = blockIdx.x * blockDim.x + threadIdx.x; i < A; i += stride) {
    float4 an = reinterpret_cast<const float4*>(anchors)[i];  // global_load_b128
    float aw = an.z - an.x;
    float ah = an.w - an.y;
    float area_a = aw * ah;
    float iw = fmaxf(fminf(an.z, bx2) - fmaxf(an.x, bx1), 0.0f);
    float ih = fmaxf(fminf(an.w, by2) - fmaxf(an.y, by1), 0.0f);
    float inter = iw * ih;
    float ua = fmaxf(area_a + area_b - inter, 1e-8f);
    bool pos = (inter / ua) >= 0.5f;
    fsum += focal_elem(pos, bd[i]);
    psum += pos ? 1.0f : 0.0f;
  }
  // all lanes reach here (no early returns) -> EXEC all 1s for WMMA
  float wf = wave_reduce_wmma(fsum);
  float wp = wave_reduce_wmma(psum);
  if ((threadIdx.x & 31) == 0) {
    atomicAdd(&ws[b], wf);
    atomicAdd(&ws[B + b], wp);
  }
}

// Classification focal: target==1 only at the (<=3) noun indices.
__global__ void cls_focal_kernel(const float* __restrict__ cls,
                                 const float* __restrict__ annotations,
                                 float* __restrict__ ws, int C, int B) {
  const int b = blockIdx.y;
  const float* ann = annotations + (size_t)b * 7;
  const int n0 = (int)ann[4], n1 = (int)ann[5], n2 = (int)ann[6];
  const float* row = cls + (size_t)b * C;

  float fsum = 0.0f;
  const int stride = gridDim.x * blockDim.x;
  for (int i = blockIdx.x * blockDim.x + threadIdx.x; i < C; i += stride) {
    bool t = (i == n0) | (i == n1) | (i == n2);
    fsum += focal_elem(t, row[i]);
  }
  float wf = wave_reduce_wmma(fsum);
  if ((threadIdx.x & 31) == 0) atomicAdd(&ws[2 * B + b], wf);
}

// One wave: per-image finalize + mean over B. The reference's reg_j[pidx]
// gathers only rows 0/1, so reg loss = (n_neg*s0 + n_pos*s1)/(4A), gated on
// n_pos > 0.
__global__ void finalize_kernel(const float* __restrict__ anchors,
                                const float* __restrict__ annotations,
                                const float* __restrict__ regressions,
                                const float* __restrict__ ws,
                                float* __restrict__ out,
                                int A, int B, int C) {
  const int b = threadIdx.x;  // 32 lanes; lanes >= B contribute zero
  float cls_l = 0.0f, bb_l = 0.0f, rg_l = 0.0f;
  if (b < B) {
    cls_l = ws[2 * B + b] / (float)C;
    bb_l  = ws[b] / (float)A;
    float npos = ws[B + b];

    const float* ann = annotations + (size_t)b * 7;
    float gx1 = ann[0], gy1 = ann[1], gx2 = ann[2], gy2 = ann[3];
    float gwraw = gx2 - gx1, ghraw = gy2 - gy1;
    float gx = gx1 + 0.5f * gwraw, gy = gy1 + 0.5f * ghraw;
    float gw = fmaxf(gwraw, 1.0f), gh = fmaxf(ghraw, 1.0f);

    float s[2];
#pragma unroll
    for (int k = 0; k < 2; ++k) {
      float x1 = anchors[4 * k + 0], y1 = anchors[4 * k + 1];
      float x2 = anchors[4 * k + 2], y2 = anchors[4 * k + 3];
      float aw = x2 - x1, ah = y2 - y1;
      float ax = x1 + 0.5f * aw, ay = y1 + 0.5f * ah;
      float t0 = ((gx - ax) / aw) * 10.0f;       // / scale 0.1
      float t1 = ((gy - ay) / ah) * 10.0f;
      float t2 = logf(gw / aw) * 5.0f;           // / scale 0.2
      float t3 = logf(gh / ah) * 5.0f;
      const float* reg = regressions + ((size_t)b * A + (size_t)k) * 4;
      s[k] = smooth_l1(fabsf(t0 - reg[0])) + smooth_l1(fabsf(t1 - reg[1])) +
             smooth_l1(fabsf(t2 - reg[2])) + smooth_l1(fabsf(t3 - reg[3]));
    }
    float nneg = (float)A - npos;
    float rl = (nneg * s[0] + npos * s[1]) / (4.0f * (float)A);
    rg_l = (npos > 0.0f) ? rl : 0.0f;
  }
  // converged: all 32 lanes execute the WMMA reductions
  const float invB = 1.0f / (float)B;
  float m_cls = wave_reduce_wmma(cls_l) * invB;
  float m_bb  = wave_reduce_wmma(bb_l)  * invB;
  float m_rg  = wave_reduce_wmma(rg_l)  * invB;
  if (threadIdx.x == 0) {
    out[0] = m_cls;
    out[1] = m_bb;
    out[2] = m_rg;
  }
}

// ---------- launch ----------

extern "C" void kernel_launch(void* const* d_in, const int* in_sizes, int n_in,
                              void* d_out, int out_size, void* d_ws, size_t ws_size,
                              hipStream_t stream) {
  const float* classifications = (const float*)d_in[0];  // (B, C)
  const float* regressions     = (const float*)d_in[1];  // (B, A, 4)
  const float* box_dist        = (const float*)d_in[2];  // (B, A)
  const float* anchors         = (const float*)d_in[3];  // (1, A, 4)
  const float* annotations     = (const float*)d_in[4];  // (B, 1, 7)

  const int A = in_sizes[3] / 4;
  const int B = in_sizes[2] / A;
  const int C = in_sizes[0] / B;

  float* ws = (float*)d_ws;
  const int nacc = 3 * B;

  zero_ws_kernel<<<(nacc + 127) / 128, 128, 0, stream>>>(ws, nacc);

  // 200 blocks * 256 threads per image ~= 4 grid-stride iterations over A=200000
  dim3 g1(200, B);
  bbox_focal_kernel<<<g1, 256, 0, stream>>>(anchors, annotations, box_dist, ws, A, B);

  // 10 blocks * 256 threads per image over C=10000
  dim3 g2(10, B);
  cls_focal_kernel<<<g2, 256, 0, stream>>>(classifications, annotations, ws, C, B);

  finalize_kernel<<<1, 32, 0, stream>>>(anchors, annotations, regressions, ws,
                                        (float*)d_out, A, B, C);
}